// VNSnowflakeDeconvBlock_58763742544105
// MI455X (gfx1250) — compile-verified
//
#include <hip/hip_runtime.h>
#include <hip/hip_bf16.h>
#include <math.h>
#include <stdint.h>

typedef __attribute__((ext_vector_type(2))) float v2f;
typedef __attribute__((ext_vector_type(8))) float v8f;
typedef unsigned int u32x4 __attribute__((ext_vector_type(4)));
typedef int          i32x8 __attribute__((ext_vector_type(8)));
typedef int          i32x4 __attribute__((ext_vector_type(4)));

#define K_NN  16
#define R_UP  2
#define EPSF  1e-8f

#if __has_builtin(__builtin_amdgcn_tensor_load_to_lds)
#define HAVE_TDM 1
#endif
#if __has_builtin(__builtin_amdgcn_global_load_async_to_lds_b32)
#define HAVE_ASYNC 1
#endif

// LDS pad mapping matching D# pad_interval=64DW / pad_amount=1DW
__device__ __forceinline__ int lpad(int q) { return q + (q >> 6); }

// ---------------------------------------------------------------------------
// KNN: one thread per query point, LDS-tiled candidate loop, register
// insertion network for the running top-16 (fully unrolled -> no scratch).
// ---------------------------------------------------------------------------
__global__ void knn_kernel(const float* __restrict__ x, int* __restrict__ idx, int N)
{
    const int b = blockIdx.y;
    const int i = blockIdx.x * blockDim.x + threadIdx.x;
    const float* xb = x + (size_t)b * N * 3;

    const float qx = xb[i*3+0], qy = xb[i*3+1], qz = xb[i*3+2];

    float bestd[K_NN];
    int   besti[K_NN];
#pragma unroll
    for (int t = 0; t < K_NN; ++t) { bestd[t] = __builtin_inff(); besti[t] = 0; }

    __shared__ float tile[256*3];

    for (int j0 = 0; j0 < N; j0 += 256) {
        if (j0 + 256 < N)   // hint next tile into cache (global_prefetch_b8)
            __builtin_prefetch(xb + (size_t)(j0 + 256)*3 + threadIdx.x*3, 0, 1);
        __syncthreads();
        for (int e = threadIdx.x; e < 256*3; e += blockDim.x)
            tile[e] = xb[(size_t)j0*3 + e];
        __syncthreads();

        for (int jj = 0; jj < 256; ++jj) {
            const int j = j0 + jj;
            const float dx = qx - tile[jj*3+0];
            const float dy = qy - tile[jj*3+1];
            const float dz = qz - tile[jj*3+2];
            const float d2 = dx*dx + dy*dy + dz*dz;
            if (j != i && d2 < bestd[K_NN-1]) {
                float dc = d2; int ic = j;
#pragma unroll
                for (int t = 0; t < K_NN; ++t) {
                    if (dc < bestd[t]) {
                        const float td = bestd[t]; const int ti = besti[t];
                        bestd[t] = dc; besti[t] = ic;
                        dc = td; ic = ti;
                    }
                }
            }
        }
    }
#pragma unroll
    for (int t = 0; t < K_NN; ++t)
        idx[((size_t)b*N + i)*K_NN + t] = besti[t];
}

// ---------------------------------------------------------------------------
// Generic vn_lin as WMMA f32 GEMM:
//   out[p][o][d] = sum_c W[o][c] * in[p][c][d],   Cout fixed at 64.
// Columns j = p*3 + d.  One block = 128 threads = 4 waves; wave w computes
// the 16x16 tile for output rows [16w, 16w+16).
//
// B staging: the 16 columns span <= 6 consecutive points -> a rectangular
// 2-D tile (6 rows x R=Cin*3 floats, row stride R).  Staged via the Tensor
// Data Mover (tensor_load_to_lds, TENSORcnt) with LDS padding 1DW/64DW to
// de-conflict the 192-DW row stride; falls back to async-to-LDS loads
// (ASYNCcnt) or plain VALU staging with the identical layout.
//
// V_WMMA_F32_16X16X4_F32 layouts (ISA 7.12.2):
//   A (16x4): lane16 -> M, VGPR r + 2*half -> K        (2 VGPRs)
//   B (4x16): VGPR r + 2*half -> K, lane16 -> N        (2 VGPRs, mirrored)
//   C/D (16x16): VGPR r -> M = r + 8*half, lane16 -> N (8 VGPRs)
// ---------------------------------------------------------------------------
__global__ void vn_lin_wmma_kernel(const float* __restrict__ in,   // [P][Cin][3]
                                   const float* __restrict__ W,    // [64][Cin]
                                   float* __restrict__ out,        // [P][64][3]
                                   int P, int Cin)
{
    const int tid  = threadIdx.x;
    const int wave = tid >> 5;
    const int lane = tid & 31;
    const int l16  = lane & 15;
    const int half = lane >> 4;
    const int j0   = blockIdx.x * 16;
    const int R    = Cin * 3;        // floats per point row
    const int p0   = j0 / 3;         // first point touched by this block

    __shared__ float Bt[1216];       // 6*192 + pad (1 DW per 64 DW)

#if defined(HAVE_TDM)
    if (wave == 0) {
        const uint64_t gaddr = (uint64_t)(uintptr_t)(in + (size_t)p0 * R);
        const uint32_t laddr = (uint32_t)(uintptr_t)&Bt[0];   // LDS aperture low bits
        u32x4 g0;
        g0[0] = 1u;                                  // count=1, user descriptor
        g0[1] = laddr;                               // lds_addr
        g0[2] = (uint32_t)gaddr;                     // global_addr[31:0]
        g0[3] = (uint32_t)(gaddr >> 32) | 0x80000000u; // global_addr[56:32] | type=2
        i32x8 g1;
        g1[0] = (int)((2u << 16)      // data_size = 4B
                    | (1u << 20)      // pad_enable
                    | (5u << 22));    // pad_interval = 64 DW (pad_amount=0 -> 1 DW)
        g1[1] = (int)(((uint32_t)R & 0xFFFFu) << 16);        // tensor_dim0[15:0]
        g1[2] = (int)(((uint32_t)P & 0xFFFFu) << 16);        // dim0 hi=0, tensor_dim1 lo
        g1[3] = (int)(((uint32_t)R << 16) | ((uint32_t)P >> 16)); // dim1 hi, tile_dim0=R
        g1[4] = 6;                                           // tile_dim1 = 6 rows
        g1[5] = R;                                           // tensor_dim0_stride
        g1[6] = 0;
        g1[7] = 0;
        i32x4 z4 = {0, 0, 0, 0};
        i32x8 z8 = {0, 0, 0, 0, 0, 0, 0, 0};
        __builtin_amdgcn_tensor_load_to_lds(g0, g1, z4, z4, z8, 0);
        __builtin_amdgcn_s_wait_tensorcnt(0);
    }
#elif defined(HAVE_ASYNC)
    typedef __attribute__((address_space(1))) void gvoid;
    typedef __attribute__((address_space(3))) void lvoid;
    for (int e = tid; e < 6 * R; e += blockDim.x) {
        const int pp  = e / R;
        const int rem = e - pp * R;
        if (p0 + pp < P) {
            const float* g = in + (size_t)(p0 + pp) * R + rem;
            __builtin_amdgcn_global_load_async_to_lds_b32(
                (gvoid*)(uintptr_t)g,
                (lvoid*)(uint32_t)(uintptr_t)&Bt[lpad(e)],
                0, 0);
        }
    }
#if __has_builtin(__builtin_amdgcn_s_wait_asynccnt)
    __builtin_amdgcn_s_wait_asynccnt(0);
#else
    asm volatile("s_wait_asynccnt 0x0" ::: "memory");
#endif
#else
    for (int e = tid; e < 6 * R; e += blockDim.x) {
        const int pp  = e / R;
        const int rem = e - pp * R;
        if (p0 + pp < P)
            Bt[lpad(e)] = in[(size_t)(p0 + pp) * R + rem];
    }
#endif
    __syncthreads();

    // per-lane column -> (point, dim) and base offset inside the LDS tile
    const int j  = j0 + l16;
    const int p  = j / 3;
    const int dd = j - p*3;
    const int colq = (p - p0) * R + dd;

    const int o0 = wave * 16;
    v8f acc = {};
#pragma unroll 4
    for (int k0 = 0; k0 < Cin; k0 += 4) {
        v2f a, bb;
        const float* wr = W + (size_t)(o0 + l16)*Cin + k0 + 2*half;
        a.x  = wr[0];                         // K = 2*half + 0
        a.y  = wr[1];                         // K = 2*half + 1
        const int q0 = colq + (k0 + 2*half) * 3;
        const int q1 = q0 + 3;
        bb.x = Bt[lpad(q0)];
        bb.y = Bt[lpad(q1)];
        acc = __builtin_amdgcn_wmma_f32_16x16x4_f32(
                  false, a, false, bb, (short)0, acc, false, false);
    }

#pragma unroll
    for (int r = 0; r < 8; ++r) {
        const int o = o0 + 8*half + r;
        out[((size_t)p*64 + o)*3 + dd] = acc[r];
    }
}

// ---------------------------------------------------------------------------
// qn[p] = sum_c ||Q[p,c,:]||,  kn[p] = sum_c ||K[p,c,:]||
// ---------------------------------------------------------------------------
__global__ void norms_kernel(const float* __restrict__ Q, const float* __restrict__ Kt,
                             float* __restrict__ qn, float* __restrict__ kn, int P)
{
    const int p = blockIdx.x * blockDim.x + threadIdx.x;
    if (p >= P) return;
    float sq = 0.f, sk = 0.f;
    for (int c = 0; c < 64; ++c) {
        const float* q = Q  + ((size_t)p*64 + c)*3;
        const float* k = Kt + ((size_t)p*64 + c)*3;
        sq += sqrtf(q[0]*q[0] + q[1]*q[1] + q[2]*q[2]);
        sk += sqrtf(k[0]*k[0] + k[1]*k[1] + k[2]*k[2]);
    }
    qn[p] = sq; kn[p] = sk;
}

// ---------------------------------------------------------------------------
// Edge features + MLP + softmax + message aggregation.  One block (192
// threads = 6 waves) per point; threads 0..15 each own one neighbor edge.
// h = Q + sum_j attn_j * U[nbr_j]
// ---------------------------------------------------------------------------
__global__ void edge_attn_kernel(const float* __restrict__ x,
                                 const float* __restrict__ Q,
                                 const float* __restrict__ Kt,
                                 const float* __restrict__ U,
                                 const float* __restrict__ qn,
                                 const float* __restrict__ kn,
                                 const int*   __restrict__ idx,
                                 const float* __restrict__ e_w1, const float* __restrict__ e_b1,
                                 const float* __restrict__ e_w2, const float* __restrict__ e_b2,
                                 const float* __restrict__ e_w3, const float* __restrict__ e_b3,
                                 float* __restrict__ h, int N)
{
    const int p   = blockIdx.x;          // global point b*N+n
    const int b   = p / N;
    const int tid = threadIdx.x;         // 0..191

    __shared__ float sQ[192];
    __shared__ float slog[K_NN];
    __shared__ float sattn[K_NN];
    __shared__ int   snbr[K_NN];

    sQ[tid] = Q[(size_t)p*192 + tid];
    if (tid < K_NN) snbr[tid] = idx[(size_t)p*K_NN + tid];
    __syncthreads();

    if (tid < K_NN) {
        const size_t pj = (size_t)b*N + snbr[tid];
        const float* Kj = Kt + pj*192;
        float dot = 0.f;
        for (int e = 0; e < 192; ++e) dot += sQ[e]*Kj[e];

        const float dx = x[(size_t)p*3+0] - x[pj*3+0];
        const float dy = x[(size_t)p*3+1] - x[pj*3+1];
        const float dz = x[(size_t)p*3+2] - x[pj*3+2];
        const float dist = sqrtf(dx*dx + dy*dy + dz*dz);

        const float s0 = qn[p], s1 = kn[pj], s2 = dot, s3 = dist;

        float h1[32], h2[32];
#pragma unroll
        for (int o = 0; o < 32; ++o) {
            float a = e_w1[o*4+0]*s0 + e_w1[o*4+1]*s1
                    + e_w1[o*4+2]*s2 + e_w1[o*4+3]*s3 + e_b1[o];
            h1[o] = a / (1.f + __expf(-a));      // silu
        }
#pragma unroll
        for (int o = 0; o < 32; ++o) {
            float a = e_b2[o];
#pragma unroll
            for (int c = 0; c < 32; ++c) a += e_w2[o*32+c]*h1[c];
            h2[o] = a / (1.f + __expf(-a));
        }
        float lg = e_b3[0];
#pragma unroll
        for (int c = 0; c < 32; ++c) lg += e_w3[c]*h2[c];
        slog[tid] = lg;
    }
    __syncthreads();

    if (tid == 0) {
        float m = -__builtin_inff();
        for (int t = 0; t < K_NN; ++t) m = fmaxf(m, slog[t]);
        float s = 0.f;
        for (int t = 0; t < K_NN; ++t) { const float e = __expf(slog[t]-m); sattn[t] = e; s += e; }
        const float inv = 1.f / s;
        for (int t = 0; t < K_NN; ++t) sattn[t] *= inv;
    }
    __syncthreads();

    float m = 0.f;
#pragma unroll
    for (int t = 0; t < K_NN; ++t) {
        const size_t pj = (size_t)b*N + snbr[t];
        m += sattn[t] * U[pj*192 + tid];
    }
    h[(size_t)p*192 + tid] = sQ[tid] + m;
}

// ---------------------------------------------------------------------------
// vn_relu(RQ,RK) -> dot with d_w2 -> radial tanh -> x_up.  One thread / point.
// ---------------------------------------------------------------------------
__global__ void finalize_kernel(const float* __restrict__ x,
                                const float* __restrict__ RQ,  // [Pu][64][3]
                                const float* __restrict__ RK,
                                const float* __restrict__ w2,  // [64]
                                float* __restrict__ x_up, int Pu)
{
    const int pu = blockIdx.x * blockDim.x + threadIdx.x;
    if (pu >= Pu) return;
    const float* xq = x + (size_t)(pu >> 1) * 3;   // parent point (R_UP=2)

    float ax = 0.f, ay = 0.f, az = 0.f;
    for (int o = 0; o < 64; ++o) {
        const float* q = RQ + ((size_t)pu*64 + o)*3;
        const float* k = RK + ((size_t)pu*64 + o)*3;
        const float qx=q[0], qy=q[1], qz=q[2];
        const float kx=k[0], ky=k[1], kz=k[2];
        const float dot = qx*kx + qy*ky + qz*kz;
        float rx = qx, ry = qy, rz = qz;
        if (dot < 0.f) {
            const float knrm = fmaxf(sqrtf(kx*kx + ky*ky + kz*kz), EPSF);
            const float c = dot / (knrm*knrm);
            rx -= c*kx; ry -= c*ky; rz -= c*kz;
        }
        const float w = w2[o];
        ax += w*rx; ay += w*ry; az += w*rz;
    }
    const float n  = fmaxf(sqrtf(ax*ax + ay*ay + az*az), EPSF);
    const float sc = tanhf(n) / n;
    x_up[(size_t)pu*3+0] = xq[0] + ax*sc;
    x_up[(size_t)pu*3+1] = xq[1] + ay*sc;
    x_up[(size_t)pu*3+2] = xq[2] + az*sc;
}

// ---------------------------------------------------------------------------
extern "C" void kernel_launch(void* const* d_in, const int* in_sizes, int n_in,
                              void* d_out, int out_size, void* d_ws, size_t ws_size,
                              hipStream_t stream)
{
    const float* x    = (const float*)d_in[0];
    const float* v    = (const float*)d_in[1];
    const float* Wq   = (const float*)d_in[2];
    const float* Wk   = (const float*)d_in[3];
    const float* Wu   = (const float*)d_in[4];
    const float* e_w1 = (const float*)d_in[5];
    const float* e_b1 = (const float*)d_in[6];
    const float* e_w2 = (const float*)d_in[7];
    const float* e_b2 = (const float*)d_in[8];
    const float* e_w3 = (const float*)d_in[9];
    const float* e_b3 = (const float*)d_in[10];
    const float* heads= (const float*)d_in[11];  // (2,32,64) == [64][64] row-major
    const float* d_w1 = (const float*)d_in[12];  // (64,32)
    const float* r_qw = (const float*)d_in[13];  // (64,64)
    const float* r_kw = (const float*)d_in[14];  // (64,64)
    const float* d_w2 = (const float*)d_in[15];  // (1,64)

    const int B  = 4;
    const int N  = in_sizes[0] / (3 * B);
    const int P  = B * N;
    const int Pu = P * R_UP;

    // Workspace layout (all chunks 256B-aligned; P*192*4 is already aligned).
    char* ws = (char*)d_ws;
    size_t off = 0;
    auto alloc = [&](size_t bytes) -> void* {
        void* p = ws + off; off += (bytes + 255) & ~(size_t)255; return p;
    };
    int*   idxb = (int*)  alloc((size_t)P * K_NN * 4);
    float* Qb   = (float*)alloc((size_t)P * 192 * 4);
    float* Kb   = (float*)alloc((size_t)P * 192 * 4);   // contiguous after Qb
    float* Ub   = (float*)alloc((size_t)P * 192 * 4);
    float* Hb   = (float*)alloc((size_t)P * 192 * 4);   // contiguous after Ub
    float* qnb  = (float*)alloc((size_t)P * 4);
    float* knb  = (float*)alloc((size_t)P * 4);
    float* Db   = (float*)alloc((size_t)Pu * 192 * 4);
    float* RQb  = Qb;   // reuse Q+K region: 2*(P*192*4) == Pu*192*4
    float* RKb  = Ub;   // reuse U+H region

    float* x_up = (float*)d_out;                  // [B][N*2][3]
    float* v_up = (float*)d_out + (size_t)Pu * 3; // [B][N*2][32][3]

    // 1. KNN
    knn_kernel<<<dim3(N/256, B), 256, 0, stream>>>(x, idxb, N);

    // 2. Q/K/U projections (WMMA GEMMs, Cout=64, Cin=64, Ncols=P*3)
    vn_lin_wmma_kernel<<<P*3/16, 128, 0, stream>>>(v, Wq, Qb, P, 64);
    vn_lin_wmma_kernel<<<P*3/16, 128, 0, stream>>>(v, Wk, Kb, P, 64);
    vn_lin_wmma_kernel<<<P*3/16, 128, 0, stream>>>(v, Wu, Ub, P, 64);

    // 3. per-point channel-norm sums
    norms_kernel<<<(P+255)/256, 256, 0, stream>>>(Qb, Kb, qnb, knb, P);

    // 4. edge MLP + softmax + message -> h
    edge_attn_kernel<<<P, 192, 0, stream>>>(x, Qb, Kb, Ub, qnb, knb, idxb,
                                            e_w1, e_b1, e_w2, e_b2, e_w3, e_b3,
                                            Hb, N);

    // 5. heads up-projection: [64][64] GEMM writes v_up directly (layout match)
    vn_lin_wmma_kernel<<<P*3/16, 128, 0, stream>>>(Hb, heads, v_up, P, 64);

    // 6. d = vn_lin(v_up, d_w1)   (Cin=32, Ncols=Pu*3)
    vn_lin_wmma_kernel<<<Pu*3/16, 128, 0, stream>>>(v_up, d_w1, Db, Pu, 32);

    // 7. rq / rk for vn_relu
    vn_lin_wmma_kernel<<<Pu*3/16, 128, 0, stream>>>(Db, r_qw, RQb, Pu, 64);
    vn_lin_wmma_kernel<<<Pu*3/16, 128, 0, stream>>>(Db, r_kw, RKb, Pu, 64);

    // 8. vn_relu + d_w2 + radial tanh -> x_up
    finalize_kernel<<<(Pu+255)/256, 256, 0, stream>>>(x, RQb, RKb, d_w2, x_up, Pu);

    (void)n_in; (void)out_size; (void)ws_size;
}